// ModuleWithCifHrOp_738734375140
// MI455X (gfx1250) — compile-verified
//
#include <hip/hip_runtime.h>
#include <stdint.h>

#define F_FIELDS 17
#define HH 300
#define WW 400
#define H_F 38
#define W_F 50
#define NPTS (H_F * W_F)   // 1900
#define TH_T 30            // tile rows   (300 = 10 * 30)
#define TW_T 100           // tile cols   (400 = 4 * 100)
#define RR 13
#define NTHREADS 256

typedef uint32_t u32x4 __attribute__((ext_vector_type(4)));
typedef uint32_t u32x8 __attribute__((ext_vector_type(8)));

// Generic pointers to LDS on gfx1250 carry the LDS byte offset in addr[31:0]
// (ISA 10.2 aperture mapping), so truncation yields the AS3 offset.
__device__ __forceinline__ uint32_t lds_addr_u32(const void* p) {
  return (uint32_t)(uintptr_t)p;
}

// CDNA5 async global->LDS copy, tracked with ASYNCcnt (CDNA5 ISA).
__device__ __forceinline__ void async_g2l_b128(uint32_t lds_byte, const float* g) {
  asm volatile("global_load_async_to_lds_b128 %0, %1, off"
               :: "v"(lds_byte), "v"(g) : "memory");
}

__device__ __forceinline__ void wait_asynccnt0() {
#if __has_builtin(__builtin_amdgcn_s_wait_asynccnt)
  __builtin_amdgcn_s_wait_asynccnt(0);
#else
  asm volatile("s_wait_asynccnt 0x0" ::: "memory");
#endif
}

__global__ __launch_bounds__(NTHREADS) void cifhr_accum_kernel(
    const float* __restrict__ cifhr, const float* __restrict__ x,
    float* __restrict__ out) {
  __shared__ __align__(16) float tile[TH_T * TW_T];  // 12 KB accumulator
  __shared__ __align__(16) float pts[4][NPTS];       // conf, xr, yr, scale (30.4 KB)
  __shared__ int plist[NPTS];
  __shared__ int pcount;

  const int tid  = threadIdx.x;
  const int f    = blockIdx.z;
  const int row0 = blockIdx.y * TH_T;
  const int col0 = blockIdx.x * TW_T;

  if (tid == 0) pcount = 0;

  const float* cbase = cifhr + (size_t)f * (HH * WW) + (size_t)row0 * WW + col0;

  // ---- TDM: one tensor_load_to_lds pulls the whole 30x100 f32 tile ----
  // (stride-400 2D tile). Issued by wave 0 only: TDM ignores EXEC, so the
  // gate must be a scalar (wave-uniform) branch.
  if ((tid >> 5) == 0) {
    const uint32_t tlds = (uint32_t)__builtin_amdgcn_readfirstlane(
        (int)lds_addr_u32(&tile[0]));
    const uint64_t ga = (uint64_t)(uintptr_t)cbase;
    u32x4 g0;
    g0[0] = 1u;                                   // count=1, user descriptor
    g0[1] = tlds;                                 // lds_addr (bytes)
    g0[2] = (uint32_t)ga;                         // global_addr[31:0]
    g0[3] = (uint32_t)((ga >> 32) & 0x01FFFFFFu)  // global_addr[56:32]
            | 0x80000000u;                        // type=2 ("image")
    u32x8 g1;
    g1[0] = 2u << 16;                             // data_size=2 -> 4 bytes
    g1[1] = 0u;                                   // tensor_dim0 = 1<<20: lo16<<16
    g1[2] = 0x10u;                                // dim0 hi16 | dim1(1<<20) lo16<<16
    g1[3] = 0x10u | (100u << 16);                 // dim1 hi16 | tile_dim0=100
    g1[4] = 30u;                                  // tile_dim1=30, tile_dim2=0 (2D)
    g1[5] = 400u;                                 // tensor_dim0_stride = 400 elems
    g1[6] = 0u;                                   // stride0 hi | stride1 lo (unused)
    g1[7] = 0u;
    asm volatile("tensor_load_to_lds %0, %1" :: "s"(g0), "s"(g1) : "memory");
    __builtin_amdgcn_s_wait_tensorcnt(0);
  }

  // ---- async stage: point planes conf/xr/yr/scale (plane 3 'b' unused) ----
  const float* xf = x + (size_t)f * 5 * NPTS;
  {
    const int planes[4] = {0, 1, 2, 4};
#pragma unroll
    for (int pl = 0; pl < 4; ++pl) {
      const float* src = xf + planes[pl] * NPTS;
      for (int i = tid; i < NPTS / 4; i += NTHREADS) {
        async_g2l_b128(lds_addr_u32(&pts[pl][i * 4]), src + i * 4);
      }
    }
  }
  wait_asynccnt0();
  __syncthreads();

  // ---- phase A: compact points whose 27x27 footprint can touch this tile ----
  for (int p = tid; p < NPTS; p += NTHREADS) {
    const float v  = pts[0][p];
    const float sc = pts[3][p];
    if (v >= 0.1f && sc * 8.0f >= 0.0f) {             // v>=V_TH, scale*STRIDE>=MIN_SCALE
      const float px = pts[1][p] * 8.0f;
      const float py = pts[2][p] * 8.0f;
      const int cx = (int)rintf(px);                  // jnp.round == round-half-even
      const int cy = (int)rintf(py);
      if (cy >= row0 - RR && cy < row0 + TH_T + RR &&
          cx >= col0 - RR && cx < col0 + TW_T + RR) {
        const int idx = atomicAdd(&pcount, 1);
        plist[idx] = p;
      }
    }
  }
  __syncthreads();
  const int n = pcount;

  // ---- phase B: cooperative blob scatter with LDS float atomics (ds_add_f32) ----
  for (int j = 0; j < n; ++j) {
    const int   p  = plist[j];                        // uniform -> LDS broadcast
    const float v  = pts[0][p];
    const float px = pts[1][p] * 8.0f;
    const float py = pts[2][p] * 8.0f;
    const float sc = pts[3][p];
    const float sigma  = fmaxf(1.0f, 4.0f * sc);      // max(1, 0.5*scale*STRIDE)
    const float s2     = sigma * sigma;
    const float trunc2 = 4.0f * s2;
    const float k8     = -0.0625f / s2;               // (-0.5/s2)/8
    const float value  = v * 0.0625f;                 // v / NEIGHBORS * FACTOR
    const int cx = (int)rintf(px);
    const int cy = (int)rintf(py);
    const int y_lo = max(cy - RR, row0);
    const int y_hi = min(cy + RR, row0 + TH_T - 1);
    const int nrows = y_hi - y_lo + 1;
    if (nrows <= 0) continue;                         // uniform branch
    const int total = nrows * 27;
    for (int i = tid; i < total; i += NTHREADS) {
      const int ry = i / 27;
      const int rx = i - ry * 27;
      const int yy = y_lo + ry;
      const int xx = cx - RR + rx;
      const int lx = xx - col0;
      if ((unsigned)lx < TW_T) {                      // tile membership == inb here
        const float fdy = (float)yy - py;
        const float fdx = (float)xx - px;
        const float dy2 = fdy * fdy;
        const float dx2 = fdx * fdx;
        const float d2  = dy2 + dx2;
        if (d2 <= trunc2) {
          float g;
          if (dy2 < 0.25f && dx2 < 0.25f) {
            g = 1.0f;                                 // "nearest" override
          } else {
            float e = fmaf(d2, k8, 1.0f);             // 1 + x/8
            e = e * e; e = e * e; e = e * e;          // ^8
            g = e;
          }
          atomicAdd(&tile[(yy - row0) * TW_T + lx], value * g);
        }
      }
    }
  }
  __syncthreads();

  // ---- write out: min(acc, 1.0), vectorized float4 loads/stores ----
  const float4* tile4 = reinterpret_cast<const float4*>(tile);
  float* obase = out + (size_t)f * (HH * WW) + (size_t)row0 * WW + col0;
  for (int i = tid; i < TH_T * TW_T / 4; i += NTHREADS) {
    const int r = i / (TW_T / 4);
    const int q = i - r * (TW_T / 4);
    float4 t = tile4[i];
    t.x = fminf(t.x, 1.0f);
    t.y = fminf(t.y, 1.0f);
    t.z = fminf(t.z, 1.0f);
    t.w = fminf(t.w, 1.0f);
    *reinterpret_cast<float4*>(obase + r * WW + q * 4) = t;
  }
}

extern "C" void kernel_launch(void* const* d_in, const int* in_sizes, int n_in,
                              void* d_out, int out_size, void* d_ws, size_t ws_size,
                              hipStream_t stream) {
  (void)in_sizes; (void)n_in; (void)d_ws; (void)ws_size; (void)out_size;
  const float* cifhr = (const float*)d_in[0];   // (17,300,400) f32 (zeros, but honored)
  const float* x     = (const float*)d_in[1];   // (17,5,38,50) f32
  float* out = (float*)d_out;                   // (17,300,400) f32
  dim3 grid(WW / TW_T, HH / TH_T, F_FIELDS);    // 4 x 10 x 17 = 680 blocks
  hipLaunchKernelGGL(cifhr_accum_kernel, grid, dim3(NTHREADS), 0, stream,
                     cifhr, x, out);
}